// SingleShotInhibition_43164421324983
// MI455X (gfx1250) — compile-verified
//
#include <hip/hip_runtime.h>

typedef __attribute__((ext_vector_type(2))) float v2f;
typedef __attribute__((ext_vector_type(4))) float v4f;
typedef __attribute__((ext_vector_type(8))) float v8f;

#define CCH    128        // channels (Toeplitz dim)
#define HW     3136       // 56*56 spatial positions per image
#define NT     64         // spatial columns per workgroup tile
#define PITCH2 160        // dword pitch of one interleaved K-row-pair (128 data + 32 pad)
                          // 160 % 64 == 32 -> hi half-wave lands on banks 32..63: conflict-free b64
#define SCOPE_TAPS 27

__global__ __launch_bounds__(256)
void inhib_toeplitz_wmma(const float* __restrict__ act,
                         const float* __restrict__ filt,
                         float* __restrict__ out)
{
    __shared__ float lds_filt[CCH];                 // filter zero-extended to 128 taps
    __shared__ float lds_x[(CCH / 2) * PITCH2];     // 128 x 64 tile, K-pair interleaved

    const int t    = threadIdx.x;
    const int lane = t & 31;
    const int wave = t >> 5;          // 0..7 -> output channel block
    const int l15  = lane & 15;
    const int hi   = lane >> 4;       // 0: K offsets {0,1}; 1: K offsets {2,3}

    const int n0 = blockIdx.x * NT;   // spatial tile start
    const int b  = blockIdx.y;        // batch image
    const float* xbase = act + (size_t)b * CCH * HW;
    float*       obase = out + (size_t)b * CCH * HW;

    // ---- stage the 128x64 activation tile into registers (b128 global loads) ----
    v4f xs[8];
#pragma unroll
    for (int it = 0; it < 8; ++it) {
        int idx = it * 256 + t;           // 0..2047 float4 slots
        int r   = idx >> 4;               // channel row 0..127
        int c4  = (idx & 15) << 2;        // column 0..60 step 4
        xs[it] = *(const v4f*)(xbase + (size_t)r * HW + n0 + c4);
    }

    // ---- filter -> LDS, zero-extended so the Toeplitz lookup needs no branch ----
    if (t < CCH) lds_filt[t] = (t < SCOPE_TAPS) ? filt[t] : 0.0f;
    __syncthreads();

    // ---- build the 16x128 Toeplitz A-fragments for this wave, in VGPR pairs ----
    // tpl[i][j] = lds_filt[(j - i + 141) & 127]   (141 = 13 + 128)
    const int i = wave * 16 + l15;    // global output row held by this lane
    v2f afrag[32];
#pragma unroll
    for (int s = 0; s < 32; ++s) {
        int j0 = 4 * s + 2 * hi;
        int k0 = (j0     - i + 141) & 127;
        int k1 = (j0 + 1 - i + 141) & 127;
        v2f av;
        av.x = lds_filt[k0];
        av.y = lds_filt[k1];
        afrag[s] = av;
    }

    // ---- activation registers -> LDS, K-pair interleaved:
    //      element (k, n) lives at dword (k>>1)*PITCH2 + 2n + (k&1) ----
#pragma unroll
    for (int it = 0; it < 8; ++it) {
        int idx  = it * 256 + t;
        int r    = idx >> 4;
        int c4   = (idx & 15) << 2;
        int base = (r >> 1) * PITCH2 + 2 * c4 + (r & 1);
        lds_x[base + 0] = xs[it].x;
        lds_x[base + 2] = xs[it].y;
        lds_x[base + 4] = xs[it].z;
        lds_x[base + 6] = xs[it].w;
    }
    __syncthreads();

    // ---- main GEMM: 32 K-steps x 4 N-subtiles of v_wmma_f32_16x16x4_f32 ----
    // B fragment {X[k][n], X[k+1][n]} is one aligned b64 in the interleaved layout.
    v8f acc0 = {}, acc1 = {}, acc2 = {}, acc3 = {};
#pragma unroll
    for (int s = 0; s < 32; ++s) {
        const v2f av = afrag[s];
        const v2f* bp = (const v2f*)&lds_x[(2 * s + hi) * PITCH2 + 2 * l15];
        v2f b0 = bp[0];     // cols n0 + l15
        v2f b1 = bp[16];    // cols n0 + 16 + l15
        v2f b2 = bp[32];    // cols n0 + 32 + l15
        v2f b3 = bp[48];    // cols n0 + 48 + l15

        acc0 = __builtin_amdgcn_wmma_f32_16x16x4_f32(false, av, false, b0, (short)0, acc0, false, false);
        acc1 = __builtin_amdgcn_wmma_f32_16x16x4_f32(false, av, false, b1, (short)0, acc1, false, false);
        acc2 = __builtin_amdgcn_wmma_f32_16x16x4_f32(false, av, false, b2, (short)0, acc2, false, false);
        acc3 = __builtin_amdgcn_wmma_f32_16x16x4_f32(false, av, false, b3, (short)0, acc3, false, false);
    }

    // ---- store: C/D layout: VGPR v -> row (v + 8*hi), col = lane&15 ----
    const int i0 = wave * 16;
#pragma unroll
    for (int v = 0; v < 8; ++v) {
        int row = i0 + v + 8 * hi;
        float* orow = obase + (size_t)row * HW + n0 + l15;
        orow[0]  = acc0[v];
        orow[16] = acc1[v];
        orow[32] = acc2[v];
        orow[48] = acc3[v];
    }
}

extern "C" void kernel_launch(void* const* d_in, const int* in_sizes, int n_in,
                              void* d_out, int out_size, void* d_ws, size_t ws_size,
                              hipStream_t stream)
{
    const float* act  = (const float*)d_in[0];   // [B, 128, 56, 56] fp32
    const float* filt = (const float*)d_in[1];   // [27] fp32
    float* out = (float*)d_out;                  // [B, 128, 56, 56] fp32

    const int batch = in_sizes[0] / (CCH * HW);  // 32 for the reference shapes
    dim3 grid(HW / NT, batch);
    inhib_toeplitz_wmma<<<grid, 256, 0, stream>>>(act, filt, out);
}